// ViterbiDecoder_15453292331440
// MI455X (gfx1250) — compile-verified
//
#include <hip/hip_runtime.h>
#include <stdint.h>

typedef unsigned int uint32x4 __attribute__((ext_vector_type(4)));
typedef int          int32x8  __attribute__((ext_vector_type(8)));
typedef int          int32x4  __attribute__((ext_vector_type(4)));

#define T_STEPS 2048
#define NSYM    4096   // floats per batch row (T_STEPS * 2)

__global__ __launch_bounds__(32)
void viterbi_k7_kernel(const float* __restrict__ y, float* __restrict__ out)
{
    __shared__ float        ysh[NSYM];        // 16 KB: staged input row (via TDM)
    __shared__ float        cumsh[64];        // path metrics
    __shared__ unsigned int tbA[64][32];      // 8 KB: decisions, states 0..31
    __shared__ unsigned int tbB[64][32];      // 8 KB: decisions, states 32..63
    __shared__ unsigned int termsh;

    const int l = threadIdx.x;                // 0..31 (wave32)
    const int b = blockIdx.x;

    // ---- TDM async tensor load: stage this batch's 4096-float row into LDS ----
    {
        unsigned long long ga = (unsigned long long)(uintptr_t)(y + (size_t)b * NSYM);
        unsigned int loff = (unsigned int)(uintptr_t)(&ysh[0]);  // low 32 bits = LDS byte offset

        uint32x4 g0;
        g0.x = 1u;                                               // count=1, no gather
        g0.y = loff;                                             // lds_addr
        g0.z = (unsigned int)(ga & 0xFFFFFFFFull);               // global_addr[31:0]
        g0.w = (unsigned int)((ga >> 32) & 0x01FFFFFFull)        // global_addr[56:32]
             | (2u << 30);                                       // type=2 ("image")

        int32x8 g1;
        g1[0] = (int)(2u << 16);                                 // data_size=2 (4 bytes), wg_mask=0
        g1[1] = (int)(((unsigned)NSYM & 0xFFFFu) << 16);         // tensor_dim0[15:0]
        g1[2] = (int)((((unsigned)NSYM >> 16) & 0xFFFFu)         // tensor_dim0[31:16]
                      | (1u << 16));                             // tensor_dim1 = 1
        g1[3] = (int)(((unsigned)NSYM & 0xFFFFu) << 16);         // tile_dim0 = 4096
        g1[4] = 1;                                               // tile_dim1 = 1, tile_dim2 = 0
        g1[5] = NSYM;                                            // tensor_dim0_stride[31:0]
        g1[6] = 0;                                               // stride0[47:32], stride1[15:0]
        g1[7] = 0;

        int32x4 z4 = {0, 0, 0, 0};
#if __clang_major__ >= 23
        int32x8 z8 = {0, 0, 0, 0, 0, 0, 0, 0};
        __builtin_amdgcn_tensor_load_to_lds(g0, g1, z4, z4, z8, 0);
#else
        __builtin_amdgcn_tensor_load_to_lds(g0, g1, z4, z4, 0);
#endif
        __builtin_amdgcn_s_wait_tensorcnt((short)0);
    }

    // ---- init path metrics: state 0 = 0, others = LARGE ----
    cumsh[l]      = (l == 0) ? 0.0f : 1.0e9f;
    cumsh[l + 32] = 1.0e9f;

    // per-lane sign constants: m = bm[op(s0=2l, u=0)] = sgn0*y0 + sgn1*y1
    // g0='1011011': out0 = bit3(l)^bit2(l)^bit0(l); g1='1111001': out1 = bit4(l)^bit3(l)^bit2(l)
    const float sgn0 = ((l ^ (l >> 2) ^ (l >> 3)) & 1) ? -1.0f : 1.0f;
    const float sgn1 = (((l >> 2) ^ (l >> 3) ^ (l >> 4)) & 1) ? -1.0f : 1.0f;

    __syncthreads();

    // ---- forward ACS: lane l reads preds (2l, 2l+1), writes targets (l, l+32) ----
    for (int c = 0; c < 64; ++c) {
        unsigned int wA = 0u, wB = 0u;
        #pragma unroll 8
        for (int k = 0; k < 32; ++k) {
            const int t = c * 32 + k;
            const float y0 = ysh[2 * t];          // uniform LDS read (broadcast)
            const float y1 = ysh[2 * t + 1];
            const float m  = sgn0 * y0 + sgn1 * y1;

            const float pe = cumsh[2 * l];        // pred even
            const float po = cumsh[2 * l + 1];    // pred odd
            const float a0 = pe + m, a1 = po - m; // target l     (u=0): bm, -bm
            const float b0 = pe - m, b1 = po + m; // target l+32  (u=1): -bm, bm

            const unsigned int dA = (a1 < a0) ? 1u : 0u;  // tie -> even pred (argmin j=0)
            const unsigned int dB = (b1 < b0) ? 1u : 0u;
            const float nA = fminf(a0, a1);
            const float nB = fminf(b0, b1);

            wA = (wA << 1) | dA;                  // step k lands at bit (31-k)
            wB = (wB << 1) | dB;

            cumsh[l]      = nA;
            cumsh[l + 32] = nB;
            __syncthreads();                      // order store(t) before load(t+1)
        }
        tbA[c][l] = wA;
        tbB[c][l] = wB;
    }

    // ---- argmin over terminal metrics (first-index tie-break, like jnp.argmin) ----
    if (l == 0) {
        float best = cumsh[0];
        unsigned int bi = 0;
        for (int i = 1; i < 64; ++i) {
            float v = cumsh[i];
            if (v < best) { best = v; bi = (unsigned int)i; }
        }
        termsh = bi;
    }
    __syncthreads();

    // ---- traceback (uniform across lanes); bit(t) = state@{t+1} >> 5 ----
    unsigned int st = termsh;                     // state at time T
    unsigned int bw = 0u;
    float* orow = out + (size_t)b * T_STEPS;
    for (int t = T_STEPS - 1; t >= 0; --t) {
        bw |= (st >> 5) << (t & 31);
        if ((t & 31) == 0) {                      // chunk full: coalesced 32-wide store
            orow[t + l] = (float)((bw >> l) & 1u);
            bw = 0u;
        }
        const unsigned int w = (st < 32u) ? tbA[t >> 5][st] : tbB[t >> 5][st - 32u];
        const unsigned int d = (w >> (31 - (t & 31))) & 1u;
        st = ((st & 31u) << 1) | d;               // state at time t
    }
}

extern "C" void kernel_launch(void* const* d_in, const int* in_sizes, int n_in,
                              void* d_out, int out_size, void* d_ws, size_t ws_size,
                              hipStream_t stream)
{
    (void)n_in; (void)out_size; (void)d_ws; (void)ws_size;
    const float* y = (const float*)d_in[0];
    float* out = (float*)d_out;
    const int B = in_sizes[0] / NSYM;             // 512
    viterbi_k7_kernel<<<B, 32, 0, stream>>>(y, out);
}